// CompletePrimitiveEquations_27977416966524
// MI455X (gfx1250) — compile-verified
//
#include <hip/hip_runtime.h>
#include <math.h>

#ifndef __has_builtin
#define __has_builtin(x) 0
#endif

#if __has_builtin(__builtin_amdgcn_global_load_async_to_lds_b32)
#define HAVE_ASYNC_LDS 1
#else
#define HAVE_ASYNC_LDS 0
#endif

#define NZ 48
#define NY 256
#define NX 512
#define NXY (NX * NY)
#define NTOT (NZ * NXY)

__device__ __forceinline__ float d1f(const float* __restrict__ f, int j, int N, int s, int ci,
                                     float c, float invh) {
  if (j == 0)     return (f[ci + s] - c) * invh;
  if (j == N - 1) return (c - f[ci - s]) * invh;
  return (f[ci + s] - f[ci - s]) * (0.5f * invh);
}

// First and second torch.gradient-semantics derivatives along one axis.
// Second derivative is grad(grad(x)) expanded analytically (incl. edges).
__device__ __forceinline__ void d12f(const float* __restrict__ f, int j, int N, int s, int ci,
                                     float c, float invh, float& d1, float& d2) {
  const float invh2 = invh * invh;
  if (j >= 2 && j <= N - 3) {
    float m2 = f[ci - 2 * s], m1 = f[ci - s], p1 = f[ci + s], p2 = f[ci + 2 * s];
    d1 = (p1 - m1) * (0.5f * invh);
    d2 = (p2 - 2.0f * c + m2) * (0.25f * invh2);
  } else if (j == 0) {
    float p1 = f[ci + s], p2 = f[ci + 2 * s];
    d1 = (p1 - c) * invh;
    d2 = (p2 - 2.0f * p1 + c) * (0.5f * invh2);
  } else if (j == 1) {
    float m1 = f[ci - s], p1 = f[ci + s], p2 = f[ci + 2 * s];
    d1 = (p1 - m1) * (0.5f * invh);
    d2 = (p2 - 3.0f * c + 2.0f * m1) * (0.25f * invh2);
  } else if (j == N - 1) {
    float m1 = f[ci - s], m2 = f[ci - 2 * s];
    d1 = (c - m1) * invh;
    d2 = (c - 2.0f * m1 + m2) * (0.5f * invh2);
  } else {  // j == N-2
    float m2 = f[ci - 2 * s], m1 = f[ci - s], p1 = f[ci + s];
    d1 = (p1 - m1) * (0.5f * invh);
    d2 = (2.0f * p1 - 3.0f * c + m2) * (0.25f * invh2);
  }
}

__global__ __launch_bounds__(NX) void prim_eq_kernel(
    const float* __restrict__ u, const float* __restrict__ v, const float* __restrict__ w,
    const float* __restrict__ T, const float* __restrict__ q, const float* __restrict__ p,
    const float* __restrict__ rho,
    const float* __restrict__ F_u, const float* __restrict__ F_v, const float* __restrict__ F_w,
    const float* __restrict__ Q, const float* __restrict__ S,
    const float* __restrict__ nu_p, const float* __restrict__ kappa_p,
    float* __restrict__ out) {
  __shared__ float s_u[NX], s_v[NX], s_w[NX], s_T[NX], s_p[NX], s_rho[NX], s_q[NX];

  const int x = threadIdx.x;
  const int y = blockIdx.x;
  const int z = blockIdx.y;
  const int gid = (z * NY + y) * NX + x;

  // Warm L2 for the z+2 planes (gfx1250 global_prefetch_b8; speculative, safe).
  if (z + 2 < NZ) {
    __builtin_prefetch(u + gid + 2 * NXY, 0, 0);
    __builtin_prefetch(v + gid + 2 * NXY, 0, 0);
    __builtin_prefetch(w + gid + 2 * NXY, 0, 0);
    __builtin_prefetch(T + gid + 2 * NXY, 0, 0);
  }

  // Stage the full X line of the 7 stencil fields into LDS (14 KB / 320 KB WGP)
  // via the gfx1250 async-to-LDS path (ASYNCcnt-tracked DMA, no VGPR staging).
#if HAVE_ASYNC_LDS
#define ALOAD(g, sm)                                                            \
  __builtin_amdgcn_global_load_async_to_lds_b32(                                \
      (__attribute__((address_space(1))) int*)((g) + gid),                      \
      (__attribute__((address_space(3))) int*)((sm) + x), 0, 0)
  ALOAD(u, s_u);
  ALOAD(v, s_v);
  ALOAD(w, s_w);
  ALOAD(T, s_T);
  ALOAD(p, s_p);
  ALOAD(rho, s_rho);
  ALOAD(q, s_q);
#undef ALOAD
#if __has_builtin(__builtin_amdgcn_s_wait_asynccnt)
  __builtin_amdgcn_s_wait_asynccnt(0);
#else
  asm volatile("s_wait_asynccnt 0" ::: "memory");
#endif
#else
  s_u[x] = u[gid];
  s_v[x] = v[gid];
  s_w[x] = w[gid];
  s_T[x] = T[gid];
  s_p[x] = p[gid];
  s_rho[x] = rho[gid];
  s_q[x] = q[gid];
#endif
  __syncthreads();

  constexpr float INV_DX = 1.0f / 100000.0f;
  constexpr float INV_DY = 1.0f / 100000.0f;
  constexpr float INV_DZ = 1.0f / 500.0f;
  constexpr float OMEGA = 7.2921e-5f;
  constexpr float GRAVITY = 9.80665f;
  constexpr float RD = 287.0f;
  constexpr float CP = 1004.0f;
  constexpr float PI = 3.14159265358979323846f;

  const float nu = nu_p[0];
  const float kappa = kappa_p[0];

  const float c_u = s_u[x], c_v = s_v[x], c_w = s_w[x], c_T = s_T[x];
  const float c_p = s_p[x], c_rho = s_rho[x], c_q = s_q[x];

  // ---- X derivatives (all from LDS; X axis fully resident) ----
  float du_dx, u_xx, dv_dx, v_xx, dw_dx, w_xx, dT_dx, T_xx;
  d12f(s_u, x, NX, 1, x, c_u, INV_DX, du_dx, u_xx);
  d12f(s_v, x, NX, 1, x, c_v, INV_DX, dv_dx, v_xx);
  d12f(s_w, x, NX, 1, x, c_w, INV_DX, dw_dx, w_xx);
  d12f(s_T, x, NX, 1, x, c_T, INV_DX, dT_dx, T_xx);
  const float dp_dx = d1f(s_p, x, NX, 1, x, c_p, INV_DX);
  const float drho_dx = d1f(s_rho, x, NX, 1, x, c_rho, INV_DX);
  const float dq_dx = d1f(s_q, x, NX, 1, x, c_q, INV_DX);

  // ---- Y derivatives (global, coalesced across X lanes) ----
  float du_dy, u_yy, dv_dy, v_yy, dw_dy, w_yy, dT_dy, T_yy;
  d12f(u, y, NY, NX, gid, c_u, INV_DY, du_dy, u_yy);
  d12f(v, y, NY, NX, gid, c_v, INV_DY, dv_dy, v_yy);
  d12f(w, y, NY, NX, gid, c_w, INV_DY, dw_dy, w_yy);
  d12f(T, y, NY, NX, gid, c_T, INV_DY, dT_dy, T_yy);
  const float dp_dy = d1f(p, y, NY, NX, gid, c_p, INV_DY);
  const float drho_dy = d1f(rho, y, NY, NX, gid, c_rho, INV_DY);
  const float dq_dy = d1f(q, y, NY, NX, gid, c_q, INV_DY);

  // ---- Z derivatives (global, coalesced) ----
  float du_dz, u_zz, dv_dz, v_zz, dw_dz, w_zz, dT_dz, T_zz;
  d12f(u, z, NZ, NXY, gid, c_u, INV_DZ, du_dz, u_zz);
  d12f(v, z, NZ, NXY, gid, c_v, INV_DZ, dv_dz, v_zz);
  d12f(w, z, NZ, NXY, gid, c_w, INV_DZ, dw_dz, w_zz);
  d12f(T, z, NZ, NXY, gid, c_T, INV_DZ, dT_dz, T_zz);
  const float dp_dz = d1f(p, z, NZ, NXY, gid, c_p, INV_DZ);
  const float drho_dz = d1f(rho, z, NZ, NXY, gid, c_rho, INV_DZ);
  const float dq_dz = d1f(q, z, NZ, NXY, gid, c_q, INV_DZ);

  // ---- point forcings ----
  const float Fu = F_u[gid], Fv = F_v[gid], Fw = F_w[gid];
  const float Qv = Q[gid], Sv = S[gid];

  // Coriolis: f = 2*Omega*sin(linspace(-pi/2, pi/2, NY)[y])
  const float lat = -0.5f * PI + (float)y * (PI / (float)(NY - 1));
  const float fcor = 2.0f * OMEGA * sinf(lat);

  // ---- tendencies ----
  const float u_adv = c_u * du_dx + c_v * du_dy + c_w * du_dz;
  const float v_adv = c_u * dv_dx + c_v * dv_dy + c_w * dv_dz;
  const float w_adv = c_u * dw_dx + c_v * dw_dy + c_w * dw_dz;

  const float visc_u = nu * (u_xx + u_yy + u_zz);
  const float visc_v = nu * (v_xx + v_yy + v_zz);
  const float visc_w = nu * (w_xx + w_yy + w_zz);

  const float inv_rho = 1.0f / (c_rho + 1e-10f);
  const float du_dt = -u_adv - dp_dx * inv_rho + fcor * c_v + visc_u + Fu;
  const float dv_dt = -v_adv - dp_dy * inv_rho - fcor * c_u + visc_v + Fv;
  const float dw_dt = -w_adv - dp_dz * inv_rho - GRAVITY + visc_w + Fw;

  const float T_adv = c_u * dT_dx + c_v * dT_dy + c_w * dT_dz;
  const float diff_T = kappa * (T_xx + T_yy + T_zz);
  const float adiabatic = (RD / CP) * c_T * c_w * dp_dz / (c_p + 1e-10f);
  const float dT_dt = -T_adv + diff_T + adiabatic + Qv * (1.0f / CP);

  const float q_adv = c_u * dq_dx + c_v * dq_dy + c_w * dq_dz;
  const float dq_dt = -q_adv + Sv;

  const float div = du_dx + dv_dy + dw_dz;
  const float drho_dt = -c_rho * div - c_u * drho_dx - c_v * drho_dy - c_w * drho_dz;
  const float dp_dt = RD * (c_rho * dT_dt + c_T * drho_dt);

  // Streaming (non-temporal) stores: write-once outputs must not evict the
  // stencil reuse working set from L2.
  __builtin_nontemporal_store(du_dt, out + 0 * NTOT + gid);
  __builtin_nontemporal_store(dv_dt, out + 1 * NTOT + gid);
  __builtin_nontemporal_store(dw_dt, out + 2 * NTOT + gid);
  __builtin_nontemporal_store(dT_dt, out + 3 * NTOT + gid);
  __builtin_nontemporal_store(dq_dt, out + 4 * NTOT + gid);
  __builtin_nontemporal_store(dp_dt, out + 5 * NTOT + gid);
  __builtin_nontemporal_store(drho_dt, out + 6 * NTOT + gid);
}

extern "C" void kernel_launch(void* const* d_in, const int* in_sizes, int n_in,
                              void* d_out, int out_size, void* d_ws, size_t ws_size,
                              hipStream_t stream) {
  (void)in_sizes; (void)n_in; (void)out_size; (void)d_ws; (void)ws_size;
  // setup_inputs() order: u v w T q p rho F_u F_v F_w Q S nu kappa
  const float* u = (const float*)d_in[0];
  const float* v = (const float*)d_in[1];
  const float* w = (const float*)d_in[2];
  const float* T = (const float*)d_in[3];
  const float* q = (const float*)d_in[4];
  const float* p = (const float*)d_in[5];
  const float* rho = (const float*)d_in[6];
  const float* F_u = (const float*)d_in[7];
  const float* F_v = (const float*)d_in[8];
  const float* F_w = (const float*)d_in[9];
  const float* Q = (const float*)d_in[10];
  const float* S = (const float*)d_in[11];
  const float* nu = (const float*)d_in[12];
  const float* kappa = (const float*)d_in[13];

  dim3 grid(NY, NZ, 1);   // one block per (y,z) line
  dim3 block(NX, 1, 1);   // full X axis = 512 threads = 16 wave32
  hipLaunchKernelGGL(prim_eq_kernel, grid, block, 0, stream,
                     u, v, w, T, q, p, rho, F_u, F_v, F_w, Q, S, nu, kappa,
                     (float*)d_out);
}